// CapsNet_53876069761470
// MI455X (gfx1250) — compile-verified
//
#include <hip/hip_runtime.h>

// CapsNet dynamic routing, fully fused, fp32 WMMA (16x16x4) on gfx1250.
// B=128, N=2048, Din=16, NC=32, DC=16, ROUTINGS=3.
//
// Round-2 restructure: low register pressure (no hoisted B-operands, no
// o_acc[32] array), vectorized LDS access via a permuted W layout and a
// transposed coefficient stage, per-capsule accumulation via ds_add_f32.

typedef float v2f __attribute__((ext_vector_type(2)));
typedef float v8f __attribute__((ext_vector_type(8)));

#define N_PTS   2048
#define DIN     16
#define NC      32
#define DC      16
#define NTILES  (N_PTS / 16)   // 128 row-tiles of 16 source capsules
#define WAVES   8
#define THREADS 256
#define EPS     1e-7f

__device__ __forceinline__ void wave_sync() { __builtin_amdgcn_wave_barrier(); }

__device__ __forceinline__ v8f wmma_f32(const v2f a, const v2f b, const v8f c) {
    return __builtin_amdgcn_wmma_f32_16x16x4_f32(false, a, false, b,
                                                 (short)0, c, false, false);
}

__global__ __launch_bounds__(THREADS)
void capsnet_routing_kernel(const float* __restrict__ x,   // [B, N, DIN]
                            const float* __restrict__ W,   // [DIN, NC*DC]
                            float* __restrict__ out)       // [B, NC, DC]
{
    // ---- LDS (~54 KB) ----
    // Permuted W: sWp[c*256 + d*16 + h*8 + q] = W[kperm][c*16+d],
    // kperm(h,q) = 4*(q>>1) + 2*h + (q&1). Per-lane B operand for a capsule
    // is then 8 contiguous floats -> two ds_load_b128.
    __shared__ __align__(16) float sWp[NC * DC * DIN];      // 32 KB
    // Transposed stage: sStage[wave][c][n_local]  (coef after softmax)
    __shared__ __align__(16) float sStage[WAVES][NC][16];   // 16 KB
    __shared__ __align__(16) float sV[DIN][NC];             //  2 KB
    __shared__ __align__(16) float sOpre[NC][DC];           //  2 KB
    __shared__ __align__(16) float sO[NC][DC];              //  2 KB
    __shared__ float sFac[NC];

    const int tid  = threadIdx.x;
    const int wave = tid >> 5;
    const int lane = tid & 31;
    const int col  = lane & 15;      // M for A, N for B/C
    const int h    = lane >> 4;      // half-wave (K interleave)
    const int bidx = blockIdx.x;

    const float* xb = x + (size_t)bidx * (N_PTS * DIN);

    // Build permuted W in LDS (one time).
    for (int i = tid; i < NC * DC * DIN; i += THREADS) {
        const int c = i >> 8, d = (i >> 4) & 15, hh = (i >> 3) & 1, q = i & 7;
        const int k = 4 * (q >> 1) + 2 * hh + (q & 1);
        sWp[i] = W[k * (NC * DC) + c * DC + d];
    }
    __syncthreads();

    for (int it = 0; it < 3; ++it) {
        // Preload V (B-layout) for the logit WMMAs; sV written + barriered
        // at the end of the previous iteration.
        v2f vlo[4], vhi[4];
        if (it > 0) {
#pragma unroll
            for (int g = 0; g < 4; ++g) {
                const int k0 = 4 * g + 2 * h;
                vlo[g].x = sV[k0][col];       vlo[g].y = sV[k0 + 1][col];
                vhi[g].x = sV[k0][16 + col];  vhi[g].y = sV[k0 + 1][16 + col];
            }
        }
        {
            float* p = &sOpre[0][0];
            for (int i = tid; i < NC * DC; i += THREADS) p[i] = 0.0f;
        }
        __syncthreads();

        // Each wave owns n-tiles wave, wave+8, ... (16 tiles of 16 rows).
        for (int mt = wave; mt < NTILES; mt += WAVES) {
            const int nrow = mt * 16 + col;
            v2f a[4];
#pragma unroll
            for (int g = 0; g < 4; ++g) {
                const int k0 = 4 * g + 2 * h;
                a[g].x = xb[nrow * DIN + k0];
                a[g].y = xb[nrow * DIN + k0 + 1];
            }

            if (it > 0) {
                // Logits b[n, 0..31] = x_tile @ V (two 16-wide N tiles).
                v8f blo = {}, bhi = {};
#pragma unroll
                for (int g = 0; g < 4; ++g) {
                    blo = wmma_f32(a[g], vlo[g], blo);
                    bhi = wmma_f32(a[g], vhi[g], bhi);
                }
                // Store transposed: element (n = j+8h, c) -> sStage[c][n].
                {
                    float4* s0 = (float4*)&sStage[wave][col][8 * h];
                    s0[0] = make_float4(blo[0], blo[1], blo[2], blo[3]);
                    s0[1] = make_float4(blo[4], blo[5], blo[6], blo[7]);
                    float4* s1 = (float4*)&sStage[wave][16 + col][8 * h];
                    s1[0] = make_float4(bhi[0], bhi[1], bhi[2], bhi[3]);
                    s1[1] = make_float4(bhi[4], bhi[5], bhi[6], bhi[7]);
                }
                wave_sync();
                // Softmax over 32 capsules per row n: 2 lanes per row
                // (lane = (h, col)); halves combined via shfl_xor(16).
                {
                    float vals[16];
#pragma unroll
                    for (int cc = 0; cc < 16; ++cc)
                        vals[cc] = sStage[wave][h * 16 + cc][col];
                    float m = vals[0];
#pragma unroll
                    for (int cc = 1; cc < 16; ++cc) m = fmaxf(m, vals[cc]);
                    m = fmaxf(m, __shfl_xor(m, 16, 32));
                    float s = 0.0f;
#pragma unroll
                    for (int cc = 0; cc < 16; ++cc) {
                        vals[cc] = __expf(vals[cc] - m);
                        s += vals[cc];
                    }
                    s += __shfl_xor(s, 16, 32);
                    const float r = 1.0f / s;
#pragma unroll
                    for (int cc = 0; cc < 16; ++cc)
                        sStage[wave][h * 16 + cc][col] = vals[cc] * r;
                }
                wave_sync();
            }

            // Per capsule: hat tile = x_tile @ W_c (4 chained WMMAs),
            // coefficient-weighted row reduction, one ds_add_f32.
            // Dynamic c + partial unroll keeps loads un-hoistable.
#pragma unroll 4
            for (int c = 0; c < NC; ++c) {
                const float4* wp =
                    (const float4*)&sWp[c * 256 + col * 16 + h * 8];
                const float4 w0 = wp[0];
                const float4 w1 = wp[1];
                v2f b0; b0.x = w0.x; b0.y = w0.y;
                v2f b1; b1.x = w0.z; b1.y = w0.w;
                v2f b2; b2.x = w1.x; b2.y = w1.y;
                v2f b3; b3.x = w1.z; b3.y = w1.w;
                v8f acc = {};
                acc = wmma_f32(a[0], b0, acc);
                acc = wmma_f32(a[1], b1, acc);
                acc = wmma_f32(a[2], b2, acc);
                acc = wmma_f32(a[3], b3, acc);

                float partial;
                if (it == 0) {
                    partial = 0.03125f * (acc[0] + acc[1] + acc[2] + acc[3] +
                                          acc[4] + acc[5] + acc[6] + acc[7]);
                } else {
                    const float4* cp =
                        (const float4*)&sStage[wave][c][8 * h];
                    const float4 c0 = cp[0];
                    const float4 c1 = cp[1];
                    partial = c0.x * acc[0] + c0.y * acc[1] +
                              c0.z * acc[2] + c0.w * acc[3] +
                              c1.x * acc[4] + c1.y * acc[5] +
                              c1.z * acc[6] + c1.w * acc[7];
                }
                // Lanes L and L+16 target the same address; the LDS atomic
                // unit serializes them correctly.
                atomicAdd(&sOpre[c][col], partial);
            }
        }
        __syncthreads();

        // Squash factor: (s2/(1+s2)) / sqrt(s2+eps).
        if (tid < NC) {
            float s2 = 0.0f;
#pragma unroll
            for (int d = 0; d < DC; ++d) {
                const float p = sOpre[tid][d];
                s2 = fmaf(p, p, s2);
            }
            sFac[tid] = (s2 / (1.0f + s2)) * rsqrtf(s2 + EPS);
        }
        __syncthreads();
        for (int i = tid; i < NC * DC; i += THREADS) {
            const int c = i >> 4, d = i & 15;
            sO[c][d] = sFac[c] * sOpre[c][d];
        }
        __syncthreads();

        if (it < 2) {
            // V[k][c] = sum_d W[k][c*16+d] * o[c][d], read from permuted W:
            // W[k][c*16+d] = sWp[c*256 + d*16 + hk*8 + qk].
            for (int i = tid; i < DIN * NC; i += THREADS) {
                const int k = i >> 5, c = i & 31;
                const int hk = (k >> 1) & 1;
                const int qk = 2 * (k >> 2) + (k & 1);
                float s = 0.0f;
#pragma unroll
                for (int d = 0; d < DC; ++d)
                    s = fmaf(sWp[c * 256 + d * 16 + hk * 8 + qk], sO[c][d], s);
                sV[k][c] = s;
            }
            __syncthreads();
        }
    }

    // Write o -> out[b, c, d].
    for (int i = tid; i < NC * DC; i += THREADS) {
        out[(size_t)bidx * (NC * DC) + i] = sO[i >> 4][i & 15];
    }
}

extern "C" void kernel_launch(void* const* d_in, const int* in_sizes, int n_in,
                              void* d_out, int out_size, void* d_ws, size_t ws_size,
                              hipStream_t stream) {
    (void)in_sizes; (void)n_in; (void)out_size; (void)d_ws; (void)ws_size;
    const float* x = (const float*)d_in[0];   // [128, 2048, 16]
    const float* W = (const float*)d_in[1];   // [1, 16, 512]
    float* out = (float*)d_out;               // [128, 32, 16]
    hipLaunchKernelGGL(capsnet_routing_kernel, dim3(128), dim3(THREADS), 0, stream,
                       x, W, out);
}